// Model_55877524521157
// MI455X (gfx1250) — compile-verified
//
#include <hip/hip_runtime.h>

// ---------------------------------------------------------------------------
// CDNA5 (gfx1250) stacked-LSTM implementation.
//   * fp32 WMMA (v_wmma_f32_16x16x4_f32) for both the bulk input-projection
//     GEMM (templated on K, fully unrolled, no guards) and the per-step
//     h @ W_hh^T recurrence matmul.
//   * All global accesses through explicit address_space(1) pointers so the
//     backend emits global_load/global_store (no flat ops, no exec juggling).
//   * Branch-free activations: native v_tanh_f32 where available; sigmoid
//     via the tanh half-angle identity (no divergent libm tanhf).
//   * One wave32 per 16-batch-row tile in the recurrence; h/c live in the
//     WMMA C/D accumulator layout across the whole time loop; h is re-laid
//     out into the A-matrix layout each step through a 1 KB LDS tile.
// ---------------------------------------------------------------------------

typedef __attribute__((ext_vector_type(2))) float v2f;
typedef __attribute__((ext_vector_type(4))) float v4f;
typedef __attribute__((ext_vector_type(8))) float v8f;

typedef const float __attribute__((address_space(1)))* cgfp;
typedef float       __attribute__((address_space(1)))* gfp;
typedef const v2f   __attribute__((address_space(1)))* cgv2p;
typedef const v4f   __attribute__((address_space(1)))* cgv4p;
typedef v4f         __attribute__((address_space(1)))* gv4p;

__device__ __forceinline__ cgfp gc(const float* p) {
  return (cgfp)(unsigned long long)p;
}
__device__ __forceinline__ gfp gm(float* p) {
  return (gfp)(unsigned long long)p;
}

struct GemmDir   { const float* W; const float* bias; float* XG; };
struct GemmChain { const float* A; long long ast, asb; GemmDir d[2]; };

struct RecDir    { const float* XG; const float* Whh; const float* h0;
                   const float* c0; float* hT; float* cT; int rev; int pad; };
struct RecChain  { RecDir d[2]; float* out; long long ot, ob; };

// Branch-free tanh: native v_tanh_f32 on gfx1250 if the builtin exists,
// otherwise a straight-line exp-based identity (no divergent branches).
__device__ __forceinline__ float fast_tanh(float x) {
#if __has_builtin(__builtin_amdgcn_tanhf)
  return __builtin_amdgcn_tanhf(x);
#else
  const float e = __expf(-2.0f * __builtin_fabsf(x));
  const float t = __builtin_fmaf(-2.0f, e / (1.0f + e), 1.0f);
  return __builtin_copysignf(t, x);
#endif
}

// sigmoid(x) = 0.5 * tanh(x/2) + 0.5  (exact identity)
__device__ __forceinline__ float sigm(float x) {
  return __builtin_fmaf(0.5f, fast_tanh(0.5f * x), 0.5f);
}

// ---------------------------------------------------------------------------
// xg = A @ W^T + b  for all T*B rows, both chains (grid.z) and dirs (grid.y).
// K is a compile-time multiple of 4 (25-wide inputs are zero-padded first).
// Output: raw dump of WMMA C/D accumulators, XG[tile][gate][lane][8].
// ---------------------------------------------------------------------------
template <int K>
__global__ __launch_bounds__(256) void xg_gemm_kernel(GemmChain g0,
                                                      GemmChain g1) {
  const GemmChain ch = (blockIdx.z == 0) ? g0 : g1;
  const GemmDir   dd = ch.d[blockIdx.y];
  const int wave = threadIdx.x >> 5;
  const int lane = threadIdx.x & 31;
  const int tile = blockIdx.x * 8 + wave;      // 0 .. 8191  (= T*16 tiles)
  const int t    = tile >> 4;
  const int b0   = (tile & 15) << 4;
  const int ncol = lane & 15;                  // N column / A row within tile
  const int half = lane >> 4;                  // 0: lanes 0-15, 1: lanes 16-31
  constexpr int NS = K / 4;

  // preload W^T into B-matrix registers (W is G x K, row-major)
  cgfp W = gc(dd.W);
  v2f Breg[NS][4];
#pragma unroll
  for (int ks = 0; ks < NS; ++ks)
#pragma unroll
    for (int n = 0; n < 4; ++n) {
      const int g = n * 16 + ncol;
      Breg[ks][n].x = W[g * K + ks * 4 + half];
      Breg[ks][n].y = W[g * K + ks * 4 + 2 + half];
    }

  // init accumulators with per-gate bias (C/D layout: every VGPR same N)
  cgfp bias = gc(dd.bias);
  v8f acc[4];
#pragma unroll
  for (int n = 0; n < 4; ++n) {
    const float bv = bias[n * 16 + ncol];
#pragma unroll
    for (int j = 0; j < 8; ++j) acc[n][j] = bv;
  }

  cgfp A = gc(ch.A);
  const long long arow =
      (long long)t * ch.ast + (long long)(b0 + ncol) * ch.asb;
  const int koffA = half * 2;                  // A layout K offset per half
#pragma unroll
  for (int ks = 0; ks < NS; ++ks) {
    const v2f a = *(cgv2p)(A + arow + ks * 4 + koffA);
#pragma unroll
    for (int n = 0; n < 4; ++n)
      acc[n] = __builtin_amdgcn_wmma_f32_16x16x4_f32(
          false, a, false, Breg[ks][n], (short)0, acc[n], false, false);
  }

  gfp XGp = gm(dd.XG);
#pragma unroll
  for (int n = 0; n < 4; ++n) {
    gfp p = XGp + (((long long)tile * 4 + n) * 32 + lane) * 8;
    v4f lo = {acc[n][0], acc[n][1], acc[n][2], acc[n][3]};
    v4f hi = {acc[n][4], acc[n][5], acc[n][6], acc[n][7]};
    *(gv4p)(p)     = lo;
    *(gv4p)(p + 4) = hi;
  }
}

// ---------------------------------------------------------------------------
// LSTM recurrence: one wave per (batch tile, direction, chain).
//   g = xg[t] + h @ Whh^T   (16 x v_wmma_f32_16x16x4_f32 per step)
//   c = sig(f)*c + sig(i)*tanh(g) ; h = sig(o)*tanh(c)
// ---------------------------------------------------------------------------
__global__ __launch_bounds__(32) void lstm_rec_kernel(RecChain rc0,
                                                      RecChain rc1) {
  __shared__ float hs[256];                    // 16x16 h tile for layout swap
  const RecChain ch = (blockIdx.z == 0) ? rc0 : rc1;
  const RecDir   dd = ch.d[blockIdx.y];
  const int bt   = blockIdx.x;
  const int lane = threadIdx.x;
  const int ncol = lane & 15;
  const int half = lane >> 4;
  const int mb   = half * 8;                   // C/D: M = r + 8*half
  const long long coloff = (long long)blockIdx.y * 16;

  // preload Whh^T (16x64) into B-matrix registers: 4 K-slices x 4 gate tiles
  cgfp Whh = gc(dd.Whh);
  v2f B[4][4];
#pragma unroll
  for (int ks = 0; ks < 4; ++ks)
#pragma unroll
    for (int n = 0; n < 4; ++n) {
      const int g = (n * 16 + ncol) * 16;      // Whh is (64,16) row-major
      B[ks][n].x = Whh[g + ks * 4 + half];
      B[ks][n].y = Whh[g + ks * 4 + 2 + half];
    }

  // init h, c (row-major [B][16] -> C/D layout), zeros if no initial state
  v8f h, c;
  if (dd.h0) {
    cgfp h0 = gc(dd.h0), c0 = gc(dd.c0);
#pragma unroll
    for (int r = 0; r < 8; ++r) {
      const int b = bt * 16 + mb + r;
      h[r] = h0[b * 16 + ncol];
      c[r] = c0[b * 16 + ncol];
    }
  } else {
#pragma unroll
    for (int r = 0; r < 8; ++r) { h[r] = 0.0f; c[r] = 0.0f; }
  }

  cgfp XGp  = gc(dd.XG);
  gfp  outp = gm(ch.out);
  const int koffA = half * 2;
  for (int step = 0; step < 512; ++step) {
    const int t = dd.rev ? (511 - step) : step;
    const long long tile = (long long)t * 16 + bt;

    // h (C/D layout) -> LDS [m][k]
#pragma unroll
    for (int r = 0; r < 8; ++r) hs[(mb + r) * 16 + ncol] = h[r];
    __syncthreads();

    // accumulators = precomputed xg tiles (coalesced b128 loads)
    v8f acc[4];
#pragma unroll
    for (int n = 0; n < 4; ++n) {
      cgfp p = XGp + ((tile * 4 + n) * 32 + lane) * 8;
      const v4f lo = *(cgv4p)(p);
      const v4f hi = *(cgv4p)(p + 4);
      acc[n][0] = lo[0]; acc[n][1] = lo[1]; acc[n][2] = lo[2]; acc[n][3] = lo[3];
      acc[n][4] = hi[0]; acc[n][5] = hi[1]; acc[n][6] = hi[2]; acc[n][7] = hi[3];
    }

    // g += h @ Whh^T : 4 K-slices x 4 gate tiles of 16x16x4 fp32 WMMA
#pragma unroll
    for (int ks = 0; ks < 4; ++ks) {
      v2f a;
      a.x = hs[ncol * 16 + ks * 4 + koffA];
      a.y = hs[ncol * 16 + ks * 4 + koffA + 1];
#pragma unroll
      for (int n = 0; n < 4; ++n)
        acc[n] = __builtin_amdgcn_wmma_f32_16x16x4_f32(
            false, a, false, B[ks][n], (short)0, acc[n], false, false);
    }

    // gate nonlinearities (PyTorch order i, f, g, o == gate tile 0..3)
#pragma unroll
    for (int r = 0; r < 8; ++r) {
      const float ig = sigm(acc[0][r]);
      const float fg = sigm(acc[1][r]);
      const float gg = fast_tanh(acc[2][r]);
      const float og = sigm(acc[3][r]);
      c[r] = fg * c[r] + ig * gg;
      h[r] = og * fast_tanh(c[r]);
    }

    // emit h to the layer output (strided: time-major or batch-first)
    const long long trow = (long long)t * ch.ot + coloff + ncol;
#pragma unroll
    for (int r = 0; r < 8; ++r) {
      const long long b = bt * 16 + mb + r;
      outp[trow + b * ch.ob] = h[r];
    }
    __syncthreads();                           // protect hs WAR for next step
  }

  if (dd.hT) {
    gfp hT = gm(dd.hT), cT = gm(dd.cT);
#pragma unroll
    for (int r = 0; r < 8; ++r) {
      const int b = bt * 16 + mb + r;
      hT[b * 16 + ncol] = h[r];
      cT[b * 16 + ncol] = c[r];
    }
  }
}

// ---------------------------------------------------------------------------
// Pad helpers: y (B,T,25) -> time-major (T,B,32) zero-padded; w_ih 64x25 ->
// 64x32 zero-padded. Removes all K guards from the GEMM.
// ---------------------------------------------------------------------------
__global__ __launch_bounds__(256) void pad_y_kernel(const float* y,
                                                    float* ypad) {
  const int idx = blockIdx.x * 256 + threadIdx.x;   // over 512*256*32
  const int k  = idx & 31;
  const int tb = idx >> 5;                          // t*256 + b
  const int b  = tb & 255;
  const int t  = tb >> 8;
  float v = 0.0f;
  if (k < 25) v = gc(y)[(b * 512 + t) * 25 + k];
  gm(ypad)[idx] = v;
}

__global__ __launch_bounds__(256) void pad_w_kernel(const float* w,
                                                    float* wpad) {
  const int idx = blockIdx.x * 256 + threadIdx.x;   // over 64*32
  const int k = idx & 31;
  const int g = idx >> 5;
  float v = 0.0f;
  if (k < 25) v = gc(w)[g * 25 + k];
  gm(wpad)[idx] = v;
}

// ---------------------------------------------------------------------------
// Host orchestration
// ---------------------------------------------------------------------------
extern "C" void kernel_launch(void* const* d_in, const int* in_sizes, int n_in,
                              void* d_out, int out_size, void* d_ws,
                              size_t ws_size, hipStream_t stream) {
  // ---- locate inputs: handle insertion-order vs alphabetical pytree order
  int xi, yi, pbase;
  if (in_sizes[0] >= 1000000) { xi = 0; yi = 1; pbase = 2; }          // x,y,params
  else { pbase = 0; xi = n_in - 2; yi = n_in - 1; }                   // params,x,y
  const bool leaf_sorted = (in_sizes[pbase] == 64);                   // b,whh,wih?
  const float* x = (const float*)d_in[xi];
  const float* y = (const float*)d_in[yi];

  struct DirP { const float* wih; const float* whh; const float* b; };
  static const int Ls[8] = {1, 3, 4, 4, 1, 3, 4, 4};
  static const int Ds[8] = {1, 1, 2, 2, 1, 1, 2, 2};
  DirP P[8][4][2];
  int cur = pbase;
  for (int cp = 0; cp < 8; ++cp)
    for (int l = 0; l < Ls[cp]; ++l)
      for (int dir = 0; dir < Ds[cp]; ++dir) {
        if (leaf_sorted) {
          P[cp][l][dir].b   = (const float*)d_in[cur++];
          P[cp][l][dir].whh = (const float*)d_in[cur++];
          P[cp][l][dir].wih = (const float*)d_in[cur++];
        } else {
          P[cp][l][dir].wih = (const float*)d_in[cur++];
          P[cp][l][dir].whh = (const float*)d_in[cur++];
          P[cp][l][dir].b   = (const float*)d_in[cur++];
        }
      }

  // ---- carve d_out (return order of the reference tuple)
  float* out = (float*)d_out;
  const long long S_SEQ = 512LL * 256 * 32;    // 4,194,304
  const long long S_ST1 = 256LL * 16;          // 4,096
  const long long S_ST8 = 8LL * 256 * 16;      // 32,768
  float* x2 = out;
  float* x3 = x2 + S_SEQ;
  float* h0 = x3 + S_SEQ;
  float* h2 = h0 + S_ST1;
  float* h3 = h2 + S_ST8;
  float* c0 = h3 + S_ST8;
  float* c2 = c0 + S_ST1;
  float* c3 = c2 + S_ST8;
  float* y2 = c3 + S_ST8;
  float* y3 = y2 + S_SEQ;
  float* h4 = y3 + S_SEQ;
  float* h6 = h4 + S_ST1;
  float* h7 = h6 + S_ST8;
  float* c4 = h7 + S_ST8;
  float* c6 = c4 + S_ST1;
  float* c7 = c6 + S_ST8;

  // ---- workspace: 4 xg buffers + 4 ping/pong buffers + padded W (8 KB)
  const long long XGSZ = 512LL * 256 * 64;     // floats per chain-dir (32 MiB)
  const long long PPSZ = 512LL * 256 * 32;     // floats per ping buffer
  if (ws_size < (size_t)(4 * XGSZ + 4 * PPSZ + 2048) * sizeof(float)) return;
  float* ws = (float*)d_ws;
  float* XG[2][2]; float* PP[2][2];
  for (int cchain = 0; cchain < 2; ++cchain)
    for (int d = 0; d < 2; ++d) XG[cchain][d] = ws + (cchain * 2 + d) * XGSZ;
  for (int cchain = 0; cchain < 2; ++cchain)
    for (int i = 0; i < 2; ++i)
      PP[cchain][i] = ws + 4 * XGSZ + (cchain * 2 + i) * PPSZ;
  float* Wpad = ws + 4 * XGSZ + 4 * PPSZ;
  // Ypad aliases PP[1][1]: consumed by stage-0 GEMM before the stage-0
  // recurrence writes PP[1][1] (stream-ordered, so this is safe).
  float* Ypad = PP[1][1];

  pad_y_kernel<<<dim3(16384), dim3(256), 0, stream>>>(y, Ypad);
  pad_w_kernel<<<dim3(8), dim3(256), 0, stream>>>(P[4][0][0].wih, Wpad);

  // ---- per-chain running input descriptor
  struct Cur { const float* in; long long ot, ob; int K; int pp; } cc[2];
  cc[0] = { x,    256LL * 32, 32, 32, 0 };     // x: (T,B,32) time-major
  cc[1] = { Ypad, 256LL * 32, 32, 32, 0 };     // y padded to (T,B,32)

  float* hSt[2][4]    = {{h0, nullptr, h2, h3}, {h4, nullptr, h6, h7}};
  float* cSt[2][4]    = {{c0, nullptr, c2, c3}, {c4, nullptr, c6, c7}};
  float* seqOut[2][4] = {{nullptr, nullptr, x2, x3}, {nullptr, nullptr, y2, y3}};

  for (int s = 0; s < 4; ++s) {
    const int L = Ls[s];
    const int D = Ds[s];
    for (int l = 0; l < L; ++l) {
      GemmChain gc2[2];
      RecChain  rc[2];
      const int Kcur = cc[0].K;                // chains always symmetric in K
      for (int cchain = 0; cchain < 2; ++cchain) {
        const int cp = cchain * 4 + s;
        gc2[cchain].A   = cc[cchain].in;
        gc2[cchain].ast = cc[cchain].ot;
        gc2[cchain].asb = cc[cchain].ob;
        for (int dir = 0; dir < D; ++dir) {
          const float* wih = P[cp][l][dir].wih;
          if (s == 0 && cchain == 1) wih = Wpad;           // padded 64x32
          gc2[cchain].d[dir].W    = wih;
          gc2[cchain].d[dir].bias = P[cp][l][dir].b;       // zeros if no bias
          gc2[cchain].d[dir].XG   = XG[cchain][dir];
        }
        if (D == 1) gc2[cchain].d[1] = gc2[cchain].d[0];

        // recurrence output target
        float* outp; long long ot, ob;
        const bool last = (l == L - 1) && (s >= 2);
        if (last) {
          outp = seqOut[cchain][s];
          if (cchain == 0) { ot = 256LL * 32; ob = 32; }   // x2/x3 (T,B,32)
          else             { ot = 32; ob = 512LL * 32; }   // y2/y3 (B,T,32)
        } else {
          outp = PP[cchain][cc[cchain].pp ^ 1];
          ot = 256LL * 16 * D;
          ob = 16LL * D;
        }
        rc[cchain].out = outp; rc[cchain].ot = ot; rc[cchain].ob = ob;
        for (int dir = 0; dir < D; ++dir) {
          RecDir rd;
          rd.XG  = XG[cchain][dir];
          rd.Whh = P[cp][l][dir].whh;
          rd.rev = dir;
          rd.pad = 0;
          if (s == 3) {                        // l03/l13 init from h2/c2,h6/c6
            rd.h0 = hSt[cchain][2] + (long long)(l * 2 + dir) * S_ST1;
            rd.c0 = cSt[cchain][2] + (long long)(l * 2 + dir) * S_ST1;
          } else { rd.h0 = nullptr; rd.c0 = nullptr; }
          if (s == 0)      { rd.hT = hSt[cchain][0]; rd.cT = cSt[cchain][0]; }
          else if (s >= 2) {
            rd.hT = hSt[cchain][s] + (long long)(l * D + dir) * S_ST1;
            rd.cT = cSt[cchain][s] + (long long)(l * D + dir) * S_ST1;
          } else           { rd.hT = nullptr; rd.cT = nullptr; }
          rc[cchain].d[dir] = rd;
        }
        if (D == 1) rc[cchain].d[1] = rc[cchain].d[0];

        // advance chain descriptor to this layer's output
        cc[cchain].in = outp;
        cc[cchain].ot = ot;
        cc[cchain].ob = ob;
        cc[cchain].K = 16 * D;
        if (!last) cc[cchain].pp ^= 1;
      }

      dim3 gb(1024, D, 2), tb(256);
      if (Kcur == 16)
        xg_gemm_kernel<16><<<gb, tb, 0, stream>>>(gc2[0], gc2[1]);
      else
        xg_gemm_kernel<32><<<gb, tb, 0, stream>>>(gc2[0], gc2[1]);
      dim3 gr(16, D, 2), tr(32);
      lstm_rec_kernel<<<gr, tr, 0, stream>>>(rc[0], rc[1]);
    }
  }
}